// PatchFeatureExtractor_57535381897341
// MI455X (gfx1250) — compile-verified
//
#include <hip/hip_runtime.h>

typedef float v2f __attribute__((ext_vector_type(2)));
typedef float v8f __attribute__((ext_vector_type(8)));

#define L_DIM 12
#define B_DIM 8
#define H_DIM 12
#define S_DIM 384
#define M_DIM 32
#define NL_DIM 3
#define NH_DIM 4
#define K_DIM (L_DIM * H_DIM)      /* 144 -> 36 K-steps of 4, fully unrolled */
#define SS (S_DIM * S_DIM)         /* 147456 */
#define WAVES_PER_BLOCK 8
#define BLOCK_THREADS (WAVES_PER_BLOCK * 32)
#define TILE_N 16
#define ST_PER_BLOCK (WAVES_PER_BLOCK * TILE_N)  /* 128 */

__global__ void pfe_zero_out(float* out, int n) {
    int i = blockIdx.x * blockDim.x + threadIdx.x;
    if (i < n) out[i] = 0.0f;
}

struct Stage {
    v2f a0, a1;        // A fragments (W), m-tiles 0 and 1
    v2f bf[B_DIM];     // B fragments (attn) for each batch
};

__global__ __launch_bounds__(BLOCK_THREADS)
void pfe_main(const float* __restrict__ attn,
              const float* __restrict__ refs,
              const int*   __restrict__ layer_idx,
              const int*   __restrict__ head_idx,
              float*       __restrict__ out) {
    __shared__ float    sW[M_DIM][K_DIM];       // 18432 B: W[m][l*H+h]
    __shared__ unsigned sOfs[K_DIM];            // per-k global element offset (l*B*H + h)*S*S
    __shared__ float    sScore[B_DIM * M_DIM];  // per-block z^2 partial sums

    const int tid = threadIdx.x;

    // ---- Build the 32x144 averaging matrix W from the index arrays ----
    for (int idx = tid; idx < M_DIM * K_DIM; idx += BLOCK_THREADS) {
        int m  = idx / K_DIM;
        int lh = idx - m * K_DIM;
        int l  = lh / H_DIM;
        int h  = lh - l * H_DIM;
        int cl = 0, ch = 0;
        #pragma unroll
        for (int i = 0; i < NL_DIM; ++i) cl += (layer_idx[m * NL_DIM + i] == l) ? 1 : 0;
        #pragma unroll
        for (int j = 0; j < NH_DIM; ++j) ch += (head_idx[m * NH_DIM + j] == h) ? 1 : 0;
        sW[m][lh] = (float)(cl * ch) * (1.0f / (float)(NL_DIM * NH_DIM));
    }
    for (int k = tid; k < K_DIM; k += BLOCK_THREADS) {
        int l = k / H_DIM;
        int h = k - l * H_DIM;
        sOfs[k] = (unsigned)((l * B_DIM * H_DIM + h) * SS);
    }
    for (int i = tid; i < B_DIM * M_DIM; i += BLOCK_THREADS) sScore[i] = 0.0f;
    __syncthreads();

    const int wave    = tid >> 5;
    const int lane    = tid & 31;
    const int col     = lane & 15;       // N column within the 16-wide tile
    const int half    = lane >> 4;       // lane half selects K offset (A/B VGPR layout)
    const int rowsel  = half * 2;        // VGPR0 holds K{0,2}, VGPR1 holds K{1,3}
    const int st_base = blockIdx.x * ST_PER_BLOCK + wave * TILE_N;
    const int stw     = st_base + col;   // per-lane pixel index (element units)

    // 8 batches x 2 m-tiles of 16x16 f32 accumulators (batch values co-resident per lane slot)
    v8f acc[B_DIM][2];
    #pragma unroll
    for (int b = 0; b < B_DIM; ++b) { acc[b][0] = {}; acc[b][1] = {}; }

    auto load_stage = [&](Stage& st, int ks) {
        unsigned o0 = sOfs[ks + rowsel]     + (unsigned)stw;
        unsigned o1 = sOfs[ks + rowsel + 1] + (unsigned)stw;
        st.a0.x = sW[col     ][ks + rowsel];
        st.a0.y = sW[col     ][ks + rowsel + 1];
        st.a1.x = sW[col + 16][ks + rowsel];
        st.a1.y = sW[col + 16][ks + rowsel + 1];
        #pragma unroll
        for (int b = 0; b < B_DIM; ++b) {
            // 32-bit add on the element index keeps saddr+voffset (scale_offset) form
            unsigned ob0 = o0 + (unsigned)(b * (H_DIM * SS));
            unsigned ob1 = o1 + (unsigned)(b * (H_DIM * SS));
            st.bf[b].x = __builtin_nontemporal_load(attn + ob0);
            st.bf[b].y = __builtin_nontemporal_load(attn + ob1);
        }
    };
    auto wmma_stage = [&](const Stage& st) {
        #pragma unroll
        for (int b = 0; b < B_DIM; ++b) {
            acc[b][0] = __builtin_amdgcn_wmma_f32_16x16x4_f32(
                false, st.a0, false, st.bf[b], (short)0, acc[b][0], false, false);
            acc[b][1] = __builtin_amdgcn_wmma_f32_16x16x4_f32(
                false, st.a1, false, st.bf[b], (short)0, acc[b][1], false, false);
        }
    };

    // ---- Fully unrolled K loop: no back-edge, scheduler pipelines stages freely ----
    #pragma unroll
    for (int ks = 0; ks < K_DIM; ks += 4) {
        Stage st;
        load_stage(st, ks);
        wmma_stage(st);
    }

    // ---- Fused epilogue: unbiased std over batch, z-score vs reference, sum z^2 ----
    const float invB  = 1.0f / (float)B_DIM;
    const float invBm = 1.0f / (float)(B_DIM - 1);
    #pragma unroll
    for (int mt = 0; mt < 2; ++mt) {
        #pragma unroll
        for (int r = 0; r < 8; ++r) {
            float x[B_DIM];
            #pragma unroll
            for (int b = 0; b < B_DIM; ++b) x[b] = acc[b][mt][r];

            float mean = 0.0f;
            #pragma unroll
            for (int b = 0; b < B_DIM; ++b) mean += x[b];
            mean *= invB;
            float var = 0.0f;
            #pragma unroll
            for (int b = 0; b < B_DIM; ++b) { float d = x[b] - mean; var += d * d; }
            var *= invBm;
            float sd = fmaxf(sqrtf(var), 1e-8f);
            float inv_sd = 1.0f / sd;

            // C/D layout: VGPR r -> M=r (lanes 0-15), M=r+8 (lanes 16-31)
            int m = mt * 16 + r + half * 8;
            float rv = refs[(size_t)m * SS + (size_t)stw];

            #pragma unroll
            for (int b = 0; b < B_DIM; ++b) {
                float z  = (x[b] - rv) * inv_sd;
                float zz = z * z;
                // reduce across the 16 lanes holding this m (stays within lane-half)
                zz += __shfl_xor(zz, 1);
                zz += __shfl_xor(zz, 2);
                zz += __shfl_xor(zz, 4);
                zz += __shfl_xor(zz, 8);
                if (col == 0) atomicAdd(&sScore[b * M_DIM + m], zz);
            }
        }
    }

    __syncthreads();
    const float invSS = 1.0f / (float)SS;
    for (int i = tid; i < B_DIM * M_DIM; i += BLOCK_THREADS)
        atomicAdd(&out[i], sScore[i] * invSS);   // out is [B, M] row-major = b*32+m
}

extern "C" void kernel_launch(void* const* d_in, const int* in_sizes, int n_in,
                              void* d_out, int out_size, void* d_ws, size_t ws_size,
                              hipStream_t stream) {
    const float* attn = (const float*)d_in[0];
    const float* refs = (const float*)d_in[1];
    const int*   li   = (const int*)d_in[2];
    const int*   hi   = (const int*)d_in[3];
    float*       out  = (float*)d_out;

    pfe_zero_out<<<1, 256, 0, stream>>>(out, out_size);
    pfe_main<<<dim3(SS / ST_PER_BLOCK), BLOCK_THREADS, 0, stream>>>(attn, refs, li, hi, out);
}